// EncoderLayer_22531398435405
// MI455X (gfx1250) — compile-verified
//
#include <hip/hip_runtime.h>
#include <hip/hip_bf16.h>

typedef float v2f __attribute__((ext_vector_type(2)));
typedef float v8f __attribute__((ext_vector_type(8)));

#define BDIM 4
#define LSEQ 1024
#define DMOD 1024
#define NHEAD 16
#define DHEAD 64
#define DFF 4096

// ---------------------------------------------------------------------------
// Generic fp32 WMMA GEMM:  C[m,n] = alpha * sum_k A[m,k]*B'[k,n] + bias[n]
//   transB=1: B'[k,n] = B[n*ldb + k]   (torch Linear / NT)
//   transB=0: B'[k,n] = B[k*ldb + n]   (NN)
//   addIdent: B' += delta(kGlobal==nGlobal)   (for M = table + I)
//   Batched over gridDim.z with (z/Hdiv, z%Hdiv) offsets for attention.
// Block: 128 threads (4 waves). Tile: 64(M) x 64(N), K-chunk 32.
// Each wave: one 16-row strip, 4 x 16x16 accumulators, v_wmma_f32_16x16x4_f32.
// Requires M%64==0, N%64==0, K%32==0 (true for every call here).
// ---------------------------------------------------------------------------
__global__ void __launch_bounds__(128) gemm_f32_wmma(
    const float* __restrict__ A, const float* __restrict__ Bm,
    const float* __restrict__ bias, float* __restrict__ C,
    int M, int N, int K, int lda, int ldb, int ldc,
    long long offAb, long long offAh, long long offBb, long long offBh,
    long long offCb, long long offCh, int Hdiv,
    float alpha, int transB, int addIdent, int doRelu)
{
    const int z  = blockIdx.z;
    const int zb = z / Hdiv, zh = z % Hdiv;
    A  += (long long)zb * offAb + (long long)zh * offAh;
    Bm += (long long)zb * offBb + (long long)zh * offBh;
    C  += (long long)zb * offCb + (long long)zh * offCh;

    __shared__ float As[64 * 33];   // [m][k], padded
    __shared__ float Bs[32 * 65];   // [k][n], padded

    const int tid  = threadIdx.x;
    const int lane = tid & 31;
    const int wave = tid >> 5;
    const int lm   = lane & 15;     // fragment row/col within 16
    const int kh   = lane >> 4;     // half select (K+2 for A/B, M+8 for C/D)
    const int m0   = blockIdx.y * 64;
    const int n0   = blockIdx.x * 64;

    v8f acc[4] = {};

    for (int k0 = 0; k0 < K; k0 += 32) {
        // ---- stage A chunk 64x32 (coalesced over k) ----
        #pragma unroll
        for (int i = 0; i < 16; ++i) {
            int idx = tid + i * 128;
            int m = idx >> 5, k = idx & 31;
            As[m * 33 + k] = A[(size_t)(m0 + m) * lda + (k0 + k)];
        }
        // ---- stage B chunk 32x64 into Bs[k][n] ----
        if (transB) {
            #pragma unroll
            for (int i = 0; i < 16; ++i) {
                int idx = tid + i * 128;
                int n = idx >> 5, k = idx & 31;
                float v = Bm[(size_t)(n0 + n) * ldb + (k0 + k)];
                if (addIdent && (k0 + k) == (n0 + n)) v += 1.0f;
                Bs[k * 65 + n] = v;
            }
        } else {
            #pragma unroll
            for (int i = 0; i < 16; ++i) {
                int idx = tid + i * 128;
                int k = idx >> 6, n = idx & 63;
                float v = Bm[(size_t)(k0 + k) * ldb + (n0 + n)];
                if (addIdent && (k0 + k) == (n0 + n)) v += 1.0f;
                Bs[k * 65 + n] = v;
            }
        }
        // prefetch next K-chunk (speculative, dropped if OOB translation)
        if (k0 + 32 < K) {
            __builtin_prefetch(&A[(size_t)(m0 + (tid >> 1)) * lda + k0 + 32 + (tid & 1) * 16], 0, 1);
        }
        __syncthreads();

        // ---- 8 x K=4 WMMA steps, 4 N-tiles each ----
        #pragma unroll
        for (int kk = 0; kk < 32; kk += 4) {
            v2f a;
            a.x = As[(wave * 16 + lm) * 33 + kk + 2 * kh];
            a.y = As[(wave * 16 + lm) * 33 + kk + 2 * kh + 1];
            #pragma unroll
            for (int nt = 0; nt < 4; ++nt) {
                v2f b;
                b.x = Bs[(kk + 2 * kh)     * 65 + nt * 16 + lm];
                b.y = Bs[(kk + 2 * kh + 1) * 65 + nt * 16 + lm];
                acc[nt] = __builtin_amdgcn_wmma_f32_16x16x4_f32(
                    false, a, false, b, (short)0, acc[nt], false, false);
            }
        }
        __syncthreads();
    }

    // ---- epilogue: scale, bias, relu ----
    #pragma unroll
    for (int nt = 0; nt < 4; ++nt) {
        int col = n0 + nt * 16 + lm;
        float bv = bias ? bias[col] : 0.0f;
        #pragma unroll
        for (int i = 0; i < 8; ++i) {
            int row = m0 + wave * 16 + i + 8 * kh;
            float v = acc[nt][i] * alpha + bv;
            if (doRelu) v = fmaxf(v, 0.0f);
            C[(size_t)row * ldc + col] = v;
        }
    }
}

// ---------------------------------------------------------------------------
// In-place softmax over rows of width 1024 (attn_weights).
// ---------------------------------------------------------------------------
__global__ void __launch_bounds__(256) row_softmax_1024(float* __restrict__ P)
{
    __shared__ float sm[256];
    const int tid = threadIdx.x;
    float* p = P + (size_t)blockIdx.x * 1024;

    float v[4];
    float lmax = -__builtin_inff();
    #pragma unroll
    for (int i = 0; i < 4; ++i) { v[i] = p[tid + i * 256]; lmax = fmaxf(lmax, v[i]); }

    sm[tid] = lmax; __syncthreads();
    for (int s = 128; s > 0; s >>= 1) { if (tid < s) sm[tid] = fmaxf(sm[tid], sm[tid + s]); __syncthreads(); }
    float m = sm[0]; __syncthreads();

    float lsum = 0.0f;
    #pragma unroll
    for (int i = 0; i < 4; ++i) { v[i] = __expf(v[i] - m); lsum += v[i]; }
    sm[tid] = lsum; __syncthreads();
    for (int s = 128; s > 0; s >>= 1) { if (tid < s) sm[tid] += sm[tid + s]; __syncthreads(); }
    float inv = 1.0f / sm[0];

    #pragma unroll
    for (int i = 0; i < 4; ++i) p[tid + i * 256] = v[i] * inv;
}

// ---------------------------------------------------------------------------
// z_b = attn[b,l,d] + b2f[b,l+1,d] (0 at l=L-1) + f2b[b,l-1,d] (0 at l=0)
// out[b,l,d] = x[b,l,d] + softmax_over_b(z)   (reference softmax along axis 0!)
// One thread per (l,d); loops over B=4 in registers.
// ---------------------------------------------------------------------------
__global__ void __launch_bounds__(256) shift_bsoftmax_residual(
    const float* __restrict__ attn, const float* __restrict__ b2f,
    const float* __restrict__ f2b,  const float* __restrict__ x,
    float* __restrict__ out)
{
    long long i = (long long)blockIdx.x * blockDim.x + threadIdx.x;   // over L*D
    if (i >= (long long)LSEQ * DMOD) return;
    int l = (int)(i / DMOD);
    int d = (int)(i % DMOD);

    float zv[BDIM];
    float mx = -__builtin_inff();
    #pragma unroll
    for (int b = 0; b < BDIM; ++b) {
        long long base = ((long long)b * LSEQ + l) * DMOD + d;
        float s = attn[base];
        if (l < LSEQ - 1) s += b2f[((long long)b * LSEQ + l + 1) * DMOD + d];
        if (l > 0)        s += f2b[((long long)b * LSEQ + l - 1) * DMOD + d];
        zv[b] = s; mx = fmaxf(mx, s);
    }
    float se = 0.0f;
    #pragma unroll
    for (int b = 0; b < BDIM; ++b) { zv[b] = __expf(zv[b] - mx); se += zv[b]; }
    float inv = 1.0f / se;
    #pragma unroll
    for (int b = 0; b < BDIM; ++b) {
        long long base = ((long long)b * LSEQ + l) * DMOD + d;
        out[base] = x[base] + zv[b] * inv;
    }
}

// ---------------------------------------------------------------------------
// out[row,:] = LayerNorm(a[row,:] (+ r[row,:])) * g + beta, width 1024.
// Biased variance (matches jnp.var), eps 1e-5. Safe in place (regs only).
// ---------------------------------------------------------------------------
__global__ void __launch_bounds__(256) add_layernorm_1024(
    const float* __restrict__ a, const float* __restrict__ r,
    const float* __restrict__ g, const float* __restrict__ beta,
    float* __restrict__ out)
{
    __shared__ float s1[256];
    __shared__ float s2[256];
    const int tid = threadIdx.x;
    const size_t row = blockIdx.x;
    const float* pa = a + row * 1024;
    const float* pr = r ? r + row * 1024 : nullptr;

    float v[4];
    float sum = 0.0f, sq = 0.0f;
    #pragma unroll
    for (int i = 0; i < 4; ++i) {
        float t = pa[tid + i * 256];
        if (pr) t += pr[tid + i * 256];
        v[i] = t; sum += t; sq += t * t;
    }
    s1[tid] = sum; s2[tid] = sq; __syncthreads();
    for (int s = 128; s > 0; s >>= 1) {
        if (tid < s) { s1[tid] += s1[tid + s]; s2[tid] += s2[tid + s]; }
        __syncthreads();
    }
    float mu  = s1[0] * (1.0f / 1024.0f);
    float var = s2[0] * (1.0f / 1024.0f) - mu * mu;
    float rs  = rsqrtf(var + 1e-5f);

    #pragma unroll
    for (int i = 0; i < 4; ++i) {
        int d = tid + i * 256;
        out[row * 1024 + d] = (v[i] - mu) * rs * g[d] + beta[d];
    }
}

// ---------------------------------------------------------------------------
extern "C" void kernel_launch(void* const* d_in, const int* in_sizes, int n_in,
                              void* d_out, int out_size, void* d_ws, size_t ws_size,
                              hipStream_t stream)
{
    const float* x     = (const float*)d_in[0];
    const float* Wq    = (const float*)d_in[1];
    const float* bq    = (const float*)d_in[2];
    const float* Wk    = (const float*)d_in[3];
    const float* bk    = (const float*)d_in[4];
    const float* Wv    = (const float*)d_in[5];
    const float* bv    = (const float*)d_in[6];
    const float* table = (const float*)d_in[7];
    const float* W1    = (const float*)d_in[8];
    const float* b1    = (const float*)d_in[9];
    const float* W2    = (const float*)d_in[10];
    const float* b2    = (const float*)d_in[11];
    const float* g1    = (const float*)d_in[12];
    const float* be1   = (const float*)d_in[13];
    const float* g2    = (const float*)d_in[14];
    const float* be2   = (const float*)d_in[15];

    float* out = (float*)d_out;
    const long long TOK = (long long)BDIM * LSEQ;        // 4096 rows
    const long long TD  = TOK * DMOD;                    // 4M floats

    float* x2_out = out;                                  // [B,L,D]
    float* attn_w = out + TD;                             // [B,H,L,L] (256 MB)

    float* W = (float*)d_ws;                              // scratch (floats)
    float* q    = W;                                      // 4M
    float* k    = W + 1 * TD;
    float* v    = W + 2 * TD;
    float* b2f  = W + 3 * TD;
    float* f2b  = W + 4 * TD;
    float* attn = W + 5 * TD;
    float* x1   = W + 6 * TD;
    float* hbuf = W;                                      // 16M, reuses q..b2f after consumed
    float* ybuf = W + 4 * TD;                             // reuses f2b after consumed

    dim3 gblk(128);
    const float inv_sqrt_dh = 0.125f;                     // 1/sqrt(64)

    // 1-3: q,k,v = x @ W*.T + b*      (M=4096,N=1024,K=1024, NT)
    gemm_f32_wmma<<<dim3(16, 64, 1), gblk, 0, stream>>>(x, Wq, bq, q, 4096, 1024, 1024,
        1024, 1024, 1024, 0,0,0,0,0,0, 1, 1.0f, 1, 0, 0);
    gemm_f32_wmma<<<dim3(16, 64, 1), gblk, 0, stream>>>(x, Wk, bk, k, 4096, 1024, 1024,
        1024, 1024, 1024, 0,0,0,0,0,0, 1, 1.0f, 1, 0, 0);
    gemm_f32_wmma<<<dim3(16, 64, 1), gblk, 0, stream>>>(x, Wv, bv, v, 4096, 1024, 1024,
        1024, 1024, 1024, 0,0,0,0,0,0, 1, 1.0f, 1, 0, 0);

    // 4: back2front = x @ (table + I)         (NN + identity)
    gemm_f32_wmma<<<dim3(16, 64, 1), gblk, 0, stream>>>(x, table, nullptr, b2f, 4096, 1024, 1024,
        1024, 1024, 1024, 0,0,0,0,0,0, 1, 1.0f, 0, 1, 0);
    // 5: front2back = x @ (table + I).T       (NT + identity)
    gemm_f32_wmma<<<dim3(16, 64, 1), gblk, 0, stream>>>(x, table, nullptr, f2b, 4096, 1024, 1024,
        1024, 1024, 1024, 0,0,0,0,0,0, 1, 1.0f, 1, 1, 0);

    // 6: scores[b,h] = q_h @ k_h^T / 8  -> attn_w slab   (64 batched GEMMs)
    gemm_f32_wmma<<<dim3(16, 16, BDIM * NHEAD), gblk, 0, stream>>>(
        q, k, nullptr, attn_w, 1024, 1024, DHEAD,
        1024, 1024, 1024,
        (long long)LSEQ * DMOD, DHEAD,                    // A offsets (b,h)
        (long long)LSEQ * DMOD, DHEAD,                    // B offsets (b,h)
        (long long)NHEAD * LSEQ * LSEQ, (long long)LSEQ * LSEQ,  // C offsets
        NHEAD, inv_sqrt_dh, 1, 0, 0);

    // 7: softmax rows of attn_w (these ARE the second output)
    row_softmax_1024<<<dim3(BDIM * NHEAD * LSEQ), dim3(256), 0, stream>>>(attn_w);

    // 8: attn_out[b,:,h,:] = P[b,h] @ V[b,:,h,:]         (NN, N=64)
    gemm_f32_wmma<<<dim3(1, 16, BDIM * NHEAD), gblk, 0, stream>>>(
        attn_w, v, nullptr, attn, 1024, DHEAD, 1024,
        1024, 1024, 1024,
        (long long)NHEAD * LSEQ * LSEQ, (long long)LSEQ * LSEQ,  // A offsets
        (long long)LSEQ * DMOD, DHEAD,                           // B offsets
        (long long)LSEQ * DMOD, DHEAD,                           // C offsets
        NHEAD, 1.0f, 0, 0, 0);

    // 9: shifts + batch-axis softmax + residual -> x1 (pre-LN)
    shift_bsoftmax_residual<<<dim3((LSEQ * DMOD + 255) / 256), dim3(256), 0, stream>>>(
        attn, b2f, f2b, x, x1);

    // 10: x1 = LN(x1) * g1 + beta1   (in place, register-carried)
    add_layernorm_1024<<<dim3((unsigned)TOK), dim3(256), 0, stream>>>(x1, nullptr, g1, be1, x1);

    // 11: h = relu(x1 @ W1.T + b1)   (M=4096,N=4096,K=1024)
    gemm_f32_wmma<<<dim3(64, 64, 1), gblk, 0, stream>>>(x1, W1, b1, hbuf, 4096, 4096, 1024,
        1024, 1024, 4096, 0,0,0,0,0,0, 1, 1.0f, 1, 0, 1);
    // 12: y = h @ W2.T + b2          (M=4096,N=1024,K=4096)
    gemm_f32_wmma<<<dim3(16, 64, 1), gblk, 0, stream>>>(hbuf, W2, b2, ybuf, 4096, 1024, 4096,
        4096, 4096, 1024, 0,0,0,0,0,0, 1, 1.0f, 1, 0, 0);

    // 13: x2 = LN(x1 + y) * g2 + beta2 -> first output slab
    add_layernorm_1024<<<dim3((unsigned)TOK), dim3(256), 0, stream>>>(x1, ybuf, g2, be2, x2_out);
}